// GNNEncoderV2_72954314490492
// MI455X (gfx1250) — compile-verified
//
#include <hip/hip_runtime.h>
#include <hip/hip_bf16.h>

typedef float v2f __attribute__((ext_vector_type(2)));
typedef float v8f __attribute__((ext_vector_type(8)));

#define GS_LOOP(i, n) for (long i = (long)blockIdx.x * blockDim.x + threadIdx.x; \
                           i < (n); i += (long)gridDim.x * blockDim.x)

__global__ void k_zero(float* p, long n) { GS_LOOP(i, n) p[i] = 0.f; }

// dst[(roff+r)*ldd + coff + c] = src[r*cols + c]
__global__ void k_pack(const float* __restrict__ s, int rows, int cols,
                       float* __restrict__ d, int ldd, int roff, int coff) {
  long n = (long)rows * cols;
  GS_LOOP(i, n) {
    int r = (int)(i / cols), c = (int)(i % cols);
    d[(long)(roff + r) * ldd + coff + c] = s[i];
  }
}

// in-degree (by dst) and optionally reverse in-degree (by src) as floats
__global__ void k_edge_count(const int* __restrict__ src, const int* __restrict__ dst,
                             long E, float* __restrict__ cf, float* __restrict__ cb) {
  GS_LOOP(e, E) {
    atomicAdd(&cf[dst[e]], 1.f);
    if (cb) atomicAdd(&cb[src[e]], 1.f);
  }
}

// forward: X[dst, 0..d) += x[src]; backward (boff>=0): X[src, boff..boff+d) += x[dst]
__global__ void k_scatter(const float* __restrict__ x, int d,
                          const int* __restrict__ src, const int* __restrict__ dst,
                          long E, float* __restrict__ X, int ldx, int boff) {
  long total = E * d;
  GS_LOOP(t, total) {
    long e = t / d; int c = (int)(t - e * (long)d);
    long s = src[e], q = dst[e];
    atomicAdd(&X[q * ldx + c], x[s * d + c]);
    if (boff >= 0) atomicAdd(&X[s * ldx + boff + c], x[q * d + c]);
  }
}

// divide accumulated sums by max(count,1); copy self features into soff cols
__global__ void k_scale_copy(float* __restrict__ X, int ldx, int d,
                             const float* __restrict__ cf, int foff,
                             const float* __restrict__ cb, int boff,
                             const float* __restrict__ x, int soff, long n) {
  long total = n * d;
  GS_LOOP(t, total) {
    long i = t / d; int c = (int)(t - i * (long)d);
    X[i * ldx + foff + c] *= 1.f / fmaxf(cf[i], 1.f);
    if (cb) X[i * ldx + boff + c] *= 1.f / fmaxf(cb[i], 1.f);
    X[i * ldx + soff + c] = x[i * d + c];
  }
}

// C[n,64] = A[n,K] @ B[K,64] + bias.  n % 16 == 0, K % 4 == 0.
// One wave = one 16x16 tile via V_WMMA_F32_16X16X4_F32 (full f32 precision).
__global__ void __launch_bounds__(128)
k_gemm_wmma_f32(const float* __restrict__ A, int lda, int K,
                const float* __restrict__ B, const float* __restrict__ bias,
                float* __restrict__ C) {
  const int lane = threadIdx.x & 31;
  const int wave = threadIdx.x >> 5;
  const long m0 = (long)blockIdx.x * 16;
  const int n0 = wave * 16;
  const int r  = lane & 15;          // A: row M ; B: col N
  const int kb = (lane >> 4) << 1;   // K sub-offset per lane half
  v8f acc;
#pragma unroll
  for (int i = 0; i < 8; ++i) acc[i] = 0.f;
  const float* Ar = A + (m0 + r) * (long)lda + kb;
  const float* Bc = B + n0 + r + (long)kb * 64;
  for (int k = 0; k < K; k += 4) {
    v2f a, b;
    a.x = Ar[k];
    a.y = Ar[k + 1];
    b.x = Bc[(long)k * 64];
    b.y = Bc[(long)k * 64 + 64];
    acc = __builtin_amdgcn_wmma_f32_16x16x4_f32(false, a, false, b,
                                                (short)0, acc, false, false);
  }
  const int nn = n0 + (lane & 15);
  const long mo = m0 + ((lane >> 4) << 3);
  const float bv = bias ? bias[nn] : 0.f;
#pragma unroll
  for (int i = 0; i < 8; ++i)
    C[(mo + i) * 64 + nn] = acc[i] + bv;
}

// per-column sum / sum-of-squares over [n,64] (blockDim.x == 64)
__global__ void k_bn_stats(const float* __restrict__ H, long n,
                           float* __restrict__ sums, float* __restrict__ sq) {
  int c = threadIdx.x;
  float s = 0.f, s2 = 0.f;
  for (long i = blockIdx.x; i < n; i += gridDim.x) {
    float v = H[i * 64 + c];
    s += v; s2 += v * v;
  }
  atomicAdd(&sums[c], s);
  atomicAdd(&sq[c], s2);
}

__global__ void k_bn_apply_relu(const float* __restrict__ H, float* __restrict__ O, long n,
                                const float* __restrict__ sums, const float* __restrict__ sq,
                                const float* __restrict__ g, const float* __restrict__ b) {
  long total = n * 64;
  GS_LOOP(t, total) {
    int c = (int)(t & 63);
    float mu = sums[c] / (float)n;
    float var = sq[c] / (float)n - mu * mu;
    float y = (H[t] - mu) * rsqrtf(var + 1e-5f) * g[c] + b[c];
    O[t] = fmaxf(y, 0.f);
  }
}

// graph_enc[c] = max_i relu(P[i,c]); non-negative so uint-bit atomicMax is order-preserving
__global__ void k_relu_max(const float* __restrict__ P, long n, unsigned* __restrict__ out) {
  int c = threadIdx.x;  // blockDim.x == 64
  float m = 0.f;
  for (long i = blockIdx.x; i < n; i += gridDim.x)
    m = fmaxf(m, P[i * 64 + c]);
  atomicMax(&out[c], __float_as_uint(m));
}

static inline unsigned nblk(long total, int bs) {
  long b = (total + bs - 1) / bs;
  if (b < 1) b = 1;
  if (b > (1L << 20)) b = (1L << 20);
  return (unsigned)b;
}

extern "C" void kernel_launch(void* const* d_in, const int* in_sizes, int n_in,
                              void* d_out, int out_size, void* d_ws, size_t ws_size,
                              hipStream_t stream) {
  const float* dag_x  = (const float*)d_in[0];
  const int*   dei    = (const int*)  d_in[1];
  const float* res_x  = (const float*)d_in[2];
  const int*   rei    = (const int*)  d_in[3];
  const float* d1f_Wl = (const float*)d_in[4],  *d1f_Wr = (const float*)d_in[5],  *d1f_b = (const float*)d_in[6];
  const float* d1b_Wl = (const float*)d_in[7],  *d1b_Wr = (const float*)d_in[8],  *d1b_b = (const float*)d_in[9];
  const float* dbn1_g = (const float*)d_in[10], *dbn1_b = (const float*)d_in[11];
  const float* d2f_Wl = (const float*)d_in[12], *d2f_Wr = (const float*)d_in[13], *d2f_b = (const float*)d_in[14];
  const float* d2b_Wl = (const float*)d_in[15], *d2b_Wr = (const float*)d_in[16], *d2b_b = (const float*)d_in[17];
  const float* dbn2_g = (const float*)d_in[18], *dbn2_b = (const float*)d_in[19];
  const float* proj_W = (const float*)d_in[20], *proj_b = (const float*)d_in[21];
  const float* r1_Wl  = (const float*)d_in[22], *r1_Wr  = (const float*)d_in[23], *r1_b  = (const float*)d_in[24];
  const float* rbn1_g = (const float*)d_in[25], *rbn1_b = (const float*)d_in[26];
  const float* r2_Wl  = (const float*)d_in[27], *r2_Wr  = (const float*)d_in[28], *r2_b  = (const float*)d_in[29];
  const float* rbn2_g = (const float*)d_in[30], *rbn2_b = (const float*)d_in[31];

  const long N  = in_sizes[0] / 5;
  const long E  = in_sizes[1] / 2;
  const long M  = in_sizes[2] / 2;
  const long Em = in_sizes[3] / 2;
  const int* dsrc = dei;  const int* ddst = dei + E;
  const int* rsrc = rei;  const int* rdst = rei + Em;

  // workspace carve-out (floats)
  size_t o = 0;
  float* base = (float*)d_ws;
  auto take = [&](size_t cnt) { float* p = base + o; o += cnt; return p; };
  const long NH = N > M ? N : M;
  float* X    = take((size_t)N * 192);   // X_cat (layer1 uses stride 16 inside)
  float* cntf = take((size_t)N);
  float* cntb = take((size_t)N);
  float* h    = take((size_t)N * 64);    // layer-1 output
  float* Hp   = take((size_t)NH * 64);   // pre-BN scratch (also projection out)
  float* W    = take(192 * 64);          // packed weights
  float* bia  = take(64);
  float* sums = take(64);
  float* sq   = take(64);
  float* cntr = take((size_t)M);
  float* Xr1  = take((size_t)M * 4);
  float* Xr2  = take((size_t)M * 128);
  float* rb   = take((size_t)M * 64);
  (void)ws_size; (void)n_in; (void)out_size;

  float* out_node = (float*)d_out;
  float* out_srv  = out_node + (size_t)N * 64;
  float* out_gr   = out_srv + (size_t)M * 64;

  const int BS = 256;
#define Z(p, n) k_zero<<<nblk((long)(n), BS), BS, 0, stream>>>((p), (long)(n))

  // ---------------- Phase A: DAG layer 1 (K=16: [mf(5)|mb(5)|x(5)|pad]) ----
  Z(cntf, N); Z(cntb, N); Z(X, N * 16); Z(W, 192 * 64); Z(sums, 64); Z(sq, 64);
  k_edge_count<<<nblk(E, BS), BS, 0, stream>>>(dsrc, ddst, E, cntf, cntb);
  k_scatter<<<nblk(E * 5, BS), BS, 0, stream>>>(dag_x, 5, dsrc, ddst, E, X, 16, 5);
  k_scale_copy<<<nblk(N * 5, BS), BS, 0, stream>>>(X, 16, 5, cntf, 0, cntb, 5, dag_x, 10, N);
  k_pack<<<1, 256, 0, stream>>>(d1f_Wl, 5, 32, W, 64, 0, 0);
  k_pack<<<1, 256, 0, stream>>>(d1b_Wl, 5, 32, W, 64, 5, 32);
  k_pack<<<1, 256, 0, stream>>>(d1f_Wr, 5, 32, W, 64, 10, 0);
  k_pack<<<1, 256, 0, stream>>>(d1b_Wr, 5, 32, W, 64, 10, 32);
  k_pack<<<1, 64, 0, stream>>>(d1f_b, 1, 32, bia, 64, 0, 0);
  k_pack<<<1, 64, 0, stream>>>(d1b_b, 1, 32, bia, 64, 0, 32);
  k_gemm_wmma_f32<<<(unsigned)(N / 16), 128, 0, stream>>>(X, 16, 16, W, bia, Hp);
  k_bn_stats<<<512, 64, 0, stream>>>(Hp, N, sums, sq);
  k_bn_apply_relu<<<nblk(N * 64, BS), BS, 0, stream>>>(Hp, h, N, sums, sq, dbn1_g, dbn1_b);

  // ---------------- Phase B: DAG layer 2 (K=192: [mf(64)|mb(64)|h(64)]) ----
  Z(X, N * 192); Z(W, 192 * 64); Z(sums, 64); Z(sq, 64);
  k_scatter<<<nblk(E * 64, BS), BS, 0, stream>>>(h, 64, dsrc, ddst, E, X, 192, 64);
  k_scale_copy<<<nblk(N * 64, BS), BS, 0, stream>>>(X, 192, 64, cntf, 0, cntb, 64, h, 128, N);
  k_pack<<<8, 256, 0, stream>>>(d2f_Wl, 64, 32, W, 64, 0, 0);
  k_pack<<<8, 256, 0, stream>>>(d2b_Wl, 64, 32, W, 64, 64, 32);
  k_pack<<<8, 256, 0, stream>>>(d2f_Wr, 64, 32, W, 64, 128, 0);
  k_pack<<<8, 256, 0, stream>>>(d2b_Wr, 64, 32, W, 64, 128, 32);
  k_pack<<<1, 64, 0, stream>>>(d2f_b, 1, 32, bia, 64, 0, 0);
  k_pack<<<1, 64, 0, stream>>>(d2b_b, 1, 32, bia, 64, 0, 32);
  k_gemm_wmma_f32<<<(unsigned)(N / 16), 128, 0, stream>>>(X, 192, 192, W, bia, Hp);
  k_bn_stats<<<512, 64, 0, stream>>>(Hp, N, sums, sq);
  k_bn_apply_relu<<<nblk(N * 64, BS), BS, 0, stream>>>(Hp, out_node, N, sums, sq, dbn2_g, dbn2_b);

  // ---------------- Phase C: projection + graph max -------------------------
  k_gemm_wmma_f32<<<(unsigned)(N / 16), 128, 0, stream>>>(out_node, 64, 64, proj_W, proj_b, Hp);
  Z(out_gr, 64);
  k_relu_max<<<512, 64, 0, stream>>>(Hp, N, (unsigned*)out_gr);

  // ---------------- Phase D: resource encoder ------------------------------
  Z(cntr, M); Z(Xr1, M * 4); Z(W, 192 * 64); Z(sums, 64); Z(sq, 64);
  k_edge_count<<<nblk(Em, BS), BS, 0, stream>>>(rsrc, rdst, Em, cntr, nullptr);
  k_scatter<<<nblk(Em * 2, BS), BS, 0, stream>>>(res_x, 2, rsrc, rdst, Em, Xr1, 4, -1);
  k_scale_copy<<<nblk(M * 2, BS), BS, 0, stream>>>(Xr1, 4, 2, cntr, 0, nullptr, 0, res_x, 2, M);
  k_pack<<<1, 256, 0, stream>>>(r1_Wl, 2, 64, W, 64, 0, 0);
  k_pack<<<1, 256, 0, stream>>>(r1_Wr, 2, 64, W, 64, 2, 0);
  k_pack<<<1, 64, 0, stream>>>(r1_b, 1, 64, bia, 64, 0, 0);
  k_gemm_wmma_f32<<<(unsigned)(M / 16), 128, 0, stream>>>(Xr1, 4, 4, W, bia, Hp);
  k_bn_stats<<<64, 64, 0, stream>>>(Hp, M, sums, sq);
  k_bn_apply_relu<<<nblk(M * 64, BS), BS, 0, stream>>>(Hp, rb, M, sums, sq, rbn1_g, rbn1_b);

  Z(Xr2, M * 128); Z(W, 192 * 64); Z(sums, 64); Z(sq, 64);
  k_scatter<<<nblk(Em * 64, BS), BS, 0, stream>>>(rb, 64, rsrc, rdst, Em, Xr2, 128, -1);
  k_scale_copy<<<nblk(M * 64, BS), BS, 0, stream>>>(Xr2, 128, 64, cntr, 0, nullptr, 0, rb, 64, M);
  k_pack<<<16, 256, 0, stream>>>(r2_Wl, 64, 64, W, 64, 0, 0);
  k_pack<<<16, 256, 0, stream>>>(r2_Wr, 64, 64, W, 64, 64, 0);
  k_pack<<<1, 64, 0, stream>>>(r2_b, 1, 64, bia, 64, 0, 0);
  k_gemm_wmma_f32<<<(unsigned)(M / 16), 128, 0, stream>>>(Xr2, 128, 128, W, bia, Hp);
  k_bn_stats<<<64, 64, 0, stream>>>(Hp, M, sums, sq);
  k_bn_apply_relu<<<nblk(M * 64, BS), BS, 0, stream>>>(Hp, out_srv, M, sums, sq, rbn2_g, rbn2_b);
#undef Z
}